// GCN_4724464025782
// MI455X (gfx1250) — compile-verified
//
#include <hip/hip_runtime.h>

#define F_IN 1433
#define F1   16
#define F2   7

typedef float v2f __attribute__((ext_vector_type(2)));
typedef float v8f __attribute__((ext_vector_type(8)));

// ---------------- degree / normalization ----------------

__global__ void k_deg_init(float* __restrict__ deg, int n) {
    int i = blockIdx.x * blockDim.x + threadIdx.x;
    if (i < n) deg[i] = 1.0f;  // self loop
}

__global__ void k_deg_count(const long long* __restrict__ dst,
                            float* __restrict__ deg, int E) {
    int e = blockIdx.x * blockDim.x + threadIdx.x;
    if (e < E) unsafeAtomicAdd(&deg[(int)dst[e]], 1.0f);
}

__global__ void k_dinv(float* __restrict__ deg, int n) {
    int i = blockIdx.x * blockDim.x + threadIdx.x;
    if (i < n) deg[i] = rsqrtf(deg[i]);  // deg >= 1 always
}

// ---------------- GEMM1: h1 = x @ W1 via V_WMMA_F32_16X16X4_F32 ----------------
// One wave computes a 16x16 tile (16 nodes x 16 out-features), K-loop over 1433.
// A (16x4 f32) layout: lanes0-15: VGPR0=K0,VGPR1=K1 ; lanes16-31: VGPR0=K2,VGPR1=K3.
// B (4x16 f32) mirrors A transposed; C/D: VGPR r = row r (lanes0-15) / row r+8 (lanes16-31).

__global__ __launch_bounds__(256) void k_gemm1(const float* __restrict__ x,
                                               const float* __restrict__ W1,
                                               float* __restrict__ h1, int n) {
    __shared__ float lw[F_IN * F1];  // 91,712 B: whole W1 in LDS
    for (int i = threadIdx.x; i < F_IN * F1; i += 256) lw[i] = W1[i];
    __syncthreads();

    const int wave = threadIdx.x >> 5;
    const int lane = threadIdx.x & 31;
    const int tile = blockIdx.x * 8 + wave;
    const int m0   = tile * 16;
    if (m0 >= n) return;  // wave-uniform

    const int half = lane >> 4;   // 0: K+0/K+1, 1: K+2/K+3
    const int lm   = lane & 15;

    const float* __restrict__ xr = x + (size_t)(m0 + lm) * F_IN + 2 * half;
    const float* __restrict__ bw = lw + (2 * half) * F1 + lm;

    v8f acc = {};
    int kb = 0;
#pragma unroll 2
    for (; kb + 4 <= 1432; kb += 4) {  // 358 iterations, covers K=0..1431
        v2f a = { xr[kb],      xr[kb + 1] };
        v2f b = { bw[kb * F1], bw[(kb + 1) * F1] };
        acc = __builtin_amdgcn_wmma_f32_16x16x4_f32(
            false, a, false, b, (short)0, acc, false, false);
    }
    {   // tail: kb == 1432, only k==1432 is valid
        int k0 = kb + 2 * half;
        v2f a = { (k0     < F_IN) ? xr[kb]           : 0.0f,
                  (k0 + 1 < F_IN) ? xr[kb + 1]       : 0.0f };
        v2f b = { (k0     < F_IN) ? bw[kb * F1]      : 0.0f,
                  (k0 + 1 < F_IN) ? bw[(kb + 1) * F1] : 0.0f };
        acc = __builtin_amdgcn_wmma_f32_16x16x4_f32(
            false, a, false, b, (short)0, acc, false, false);
    }

    float* __restrict__ o = h1 + (size_t)(m0 + 8 * half) * F1 + lm;
#pragma unroll
    for (int r = 0; r < 8; ++r) o[(size_t)r * F1] = acc[r];
}

// ---------------- layer-1 aggregation ----------------

__global__ void k_self1(const float* __restrict__ h1, const float* __restrict__ dinv,
                        float* __restrict__ out1, int n16) {
    int t = blockIdx.x * blockDim.x + threadIdx.x;
    if (t < n16) {
        float di = dinv[t >> 4];
        out1[t] = h1[t] * di * di;
    }
}

__global__ void k_scat1(const long long* __restrict__ src, const long long* __restrict__ dst,
                        const float* __restrict__ dinv, const float* __restrict__ h1,
                        float* __restrict__ out1, int E4) {
    int t = blockIdx.x * blockDim.x + threadIdx.x;
    if (t >= E4) return;
    int e = t >> 2, q = (t & 3) * 4;
    int s = (int)src[e], d = (int)dst[e];
    float w = dinv[s] * dinv[d];
    const float4 v = *(const float4*)(h1 + (size_t)s * F1 + q);
    float* o = out1 + (size_t)d * F1 + q;
    unsafeAtomicAdd(o + 0, v.x * w);
    unsafeAtomicAdd(o + 1, v.y * w);
    unsafeAtomicAdd(o + 2, v.z * w);
    unsafeAtomicAdd(o + 3, v.w * w);
}

// ---------------- layer 2: relu(out1+b1) @ W2, plus self-loop init of out2 ----------------

__global__ __launch_bounds__(256) void k_layer2(const float* __restrict__ out1,
                                                const float* __restrict__ b1,
                                                const float* __restrict__ W2,
                                                const float* __restrict__ dinv,
                                                float* __restrict__ h3,
                                                float* __restrict__ out2, int n) {
    __shared__ float lw2[F1 * F2];
    __shared__ float lb1[F1];
    if (threadIdx.x < F1 * F2) lw2[threadIdx.x] = W2[threadIdx.x];
    if (threadIdx.x < F1)      lb1[threadIdx.x] = b1[threadIdx.x];
    __syncthreads();

    int i = blockIdx.x * 256 + threadIdx.x;
    if (i >= n) return;

    float t[F1];
    const float* __restrict__ r = out1 + (size_t)i * F1;
#pragma unroll
    for (int k = 0; k < F1; ++k) {
        float v = r[k] + lb1[k];
        t[k] = v > 0.0f ? v : 0.0f;  // ReLU
    }
    float di = dinv[i], dd = di * di;
    float* hp = h3   + (size_t)i * 8;
    float* op = out2 + (size_t)i * 8;
#pragma unroll
    for (int j = 0; j < F2; ++j) {
        float s = 0.0f;
#pragma unroll
        for (int k = 0; k < F1; ++k) s = fmaf(t[k], lw2[k * F2 + j], s);
        hp[j] = s;
        op[j] = s * dd;  // self-loop contribution
    }
    hp[7] = 0.0f;
    op[7] = 0.0f;
}

__global__ void k_scat2(const long long* __restrict__ src, const long long* __restrict__ dst,
                        const float* __restrict__ dinv, const float* __restrict__ h3,
                        float* __restrict__ out2, int E2) {
    int t = blockIdx.x * blockDim.x + threadIdx.x;
    if (t >= E2) return;
    int e = t >> 1, q = (t & 1) * 4;
    int s = (int)src[e], d = (int)dst[e];
    float w = dinv[s] * dinv[d];
    const float4 v = *(const float4*)(h3 + (size_t)s * 8 + q);
    float* o = out2 + (size_t)d * 8 + q;
    unsafeAtomicAdd(o + 0, v.x * w);
    unsafeAtomicAdd(o + 1, v.y * w);
    unsafeAtomicAdd(o + 2, v.z * w);
    if (q == 0) unsafeAtomicAdd(o + 3, v.w * w);  // feature 7 doesn't exist
}

// ---------------- log_softmax over 7 classes ----------------

__global__ void k_lsm(const float* __restrict__ out2, const float* __restrict__ b2,
                      float* __restrict__ y, int n) {
    int i = blockIdx.x * blockDim.x + threadIdx.x;
    if (i >= n) return;
    const float* r = out2 + (size_t)i * 8;
    float v[F2];
#pragma unroll
    for (int j = 0; j < F2; ++j) v[j] = r[j] + b2[j];
    float m = v[0];
#pragma unroll
    for (int j = 1; j < F2; ++j) m = fmaxf(m, v[j]);
    float sum = 0.0f;
#pragma unroll
    for (int j = 0; j < F2; ++j) sum += expf(v[j] - m);
    float l = logf(sum);
    float* o = y + (size_t)i * F2;
#pragma unroll
    for (int j = 0; j < F2; ++j) o[j] = v[j] - m - l;
}

// ---------------- host launcher ----------------

extern "C" void kernel_launch(void* const* d_in, const int* in_sizes, int n_in,
                              void* d_out, int out_size, void* d_ws, size_t ws_size,
                              hipStream_t stream) {
    const float*     x   = (const float*)d_in[0];
    const long long* ei  = (const long long*)d_in[1];
    const float*     W1  = (const float*)d_in[2];
    const float*     b1  = (const float*)d_in[3];
    const float*     W2  = (const float*)d_in[4];
    const float*     b2  = (const float*)d_in[5];

    const int n = in_sizes[0] / F_IN;       // 50000
    const int E = in_sizes[1] / 2;          // 1,600,000
    const long long* src = ei;
    const long long* dst = ei + E;

    float* ws   = (float*)d_ws;
    float* deg  = ws;                             // n floats (becomes dinv in place)
    float* h1   = deg  + n;                       // n*16
    float* out1 = h1   + (size_t)n * F1;          // n*16
    float* h3   = out1 + (size_t)n * F1;          // n*8
    float* out2 = h3   + (size_t)n * 8;           // n*8
    float* y    = (float*)d_out;                  // n*7

    const int B = 256;
    // degrees & dinv
    k_deg_init<<<(n + B - 1) / B, B, 0, stream>>>(deg, n);
    k_deg_count<<<(E + B - 1) / B, B, 0, stream>>>(dst, deg, E);
    k_dinv<<<(n + B - 1) / B, B, 0, stream>>>(deg, n);

    // h1 = x @ W1  (WMMA, 16 rows per wave, 8 waves per block)
    int tiles = (n + 15) / 16;
    k_gemm1<<<(tiles + 7) / 8, B, 0, stream>>>(x, W1, h1, n);

    // layer-1 aggregation
    k_self1<<<((n * F1) + B - 1) / B, B, 0, stream>>>(h1, deg, out1, n * F1);
    k_scat1<<<((E * 4) + B - 1) / B, B, 0, stream>>>(src, dst, deg, h1, out1, E * 4);

    // layer 2 (bias + relu + GEMM2 + self-loop init)
    k_layer2<<<(n + B - 1) / B, B, 0, stream>>>(out1, b1, W2, deg, h3, out2, n);
    k_scat2<<<((E * 2) + B - 1) / B, B, 0, stream>>>(src, dst, deg, h3, out2, E * 2);

    // bias + log_softmax
    k_lsm<<<(n + B - 1) / B, B, 0, stream>>>(out2, b2, y, n);
}